// RadonForward_16947940950667
// MI455X (gfx1250) — compile-verified
//
#include <hip/hip_runtime.h>

typedef __attribute__((ext_vector_type(2)))  float    v2f;
typedef __attribute__((ext_vector_type(8)))  float    v8f;

#define IMG_N    256
#define NANG     180
#define NPOSN    256
#define NBATCH   4

#if __has_builtin(__builtin_amdgcn_global_load_async_to_lds_b128)
#define HAVE_ASYNC_LDS 1
#else
#define HAVE_ASYNC_LDS 0
#endif

// Gather one pixel from the LDS-staged slice with the reference's
// clamp + validity-mask semantics (out-of-range contributes 0).
__device__ __forceinline__ float fetch_lds(const float* simg, int y, int x) {
    bool valid = ((unsigned)x < IMG_N) & ((unsigned)y < IMG_N);
    int yc = min(max(y, 0), IMG_N - 1);
    int xc = min(max(x, 0), IMG_N - 1);
    float v = simg[yc * IMG_N + xc];
    return valid ? v : 0.0f;
}

__device__ __forceinline__ float sample_bilinear(const float* simg, float xi, float yi) {
    float x0 = floorf(xi), y0 = floorf(yi);
    float wx = xi - x0,    wy = yi - y0;
    int   x0i = (int)x0,   y0i = (int)y0;
    float v00 = fetch_lds(simg, y0i,     x0i);
    float v01 = fetch_lds(simg, y0i,     x0i + 1);
    float v10 = fetch_lds(simg, y0i + 1, x0i);
    float v11 = fetch_lds(simg, y0i + 1, x0i + 1);
    float owx = 1.0f - wx, owy = 1.0f - wy;
    return v00 * (owy * owx) + v01 * (owy * wx) + v10 * (wy * owx) + v11 * (wy * wx);
}

// One block = one (batch, angle) pair * half the positions.
// 8 wave32s per block; each wave produces 16 outputs (positions) by
// accumulating 256 ray samples through V_WMMA_F32_16X16X4_F32 with an
// all-ones B matrix (row-sum reduction on the matrix pipe, fp32 throughout).
// The 256 KB image slice is staged into LDS with the CDNA5 async
// global->LDS DMA path (ASYNCcnt) when available.
__global__ __launch_bounds__(256) void radon_fwd_kernel(
    const float* __restrict__ image,      // (4,1,256,256)
    const float* __restrict__ thetas,     // (180,)
    const float* __restrict__ positions,  // (256,)
    float* __restrict__ out)              // (4,1,180,256)
{
    extern __shared__ float simg[];       // 256*256*4 = 256 KB dynamic LDS

    const int pgrp = blockIdx.x & 1;                 // which 128-position half
    const int a    = (blockIdx.x >> 1) % NANG;       // angle
    const int b    =  blockIdx.x / (2 * NANG);       // batch

    // ---- Stage image slice b into LDS ----
#if HAVE_ASYNC_LDS
    {
        // Builtin signature (from compiler diagnostic): param 1 is
        // AS(1) pointer to int __vector(4); use matching vector-typed,
        // address-space-qualified pointers for both arguments.
        typedef __attribute__((__vector_size__(4 * sizeof(int)))) int v4i;
        typedef __attribute__((address_space(1))) v4i g1_v4i;
        typedef __attribute__((address_space(3))) v4i l3_v4i;
        const char* srcb = (const char*)(image + (size_t)b * IMG_N * IMG_N);
        char*       dstb = (char*)simg;
        // 256 threads x 16B async transfers; no VGPR destinations, no
        // load->ds round trip. Each thread issues 64 b128 async DMAs.
        for (int i = threadIdx.x * 16; i < IMG_N * IMG_N * 4; i += 256 * 16) {
            __builtin_amdgcn_global_load_async_to_lds_b128(
                (g1_v4i*)(srcb + i), (l3_v4i*)(dstb + i), 0, 0);
        }
#if __has_builtin(__builtin_amdgcn_s_wait_asynccnt)
        __builtin_amdgcn_s_wait_asynccnt(0);
#else
        asm volatile("s_wait_asynccnt 0x0" ::: "memory");
#endif
    }
#else
    {
        const float4* src = (const float4*)(image + (size_t)b * IMG_N * IMG_N);
        float4*       dst = (float4*)simg;
        for (int i = threadIdx.x; i < (IMG_N * IMG_N) / 4; i += blockDim.x)
            dst[i] = src[i];
    }
#endif
    __syncthreads();

    const int wave = threadIdx.x >> 5;   // wave32
    const int lane = threadIdx.x & 31;
    const int m    = lane & 15;          // A-matrix row this lane feeds
    const int h    = lane >> 4;          // lane half selects K pair / D row half

    const int   pbase = pgrp * 128 + wave * 16;
    const float s     = positions[pbase + m];
    const float th    = thetas[a];
    const float cosv  = cosf(th);
    const float sinv  = sinf(th);
    const float half  = (float)(IMG_N - 1) * 0.5f;   // 127.5
    const float xi0   = s * cosv + half;             // xi(t) = xi0 - t*sin
    const float yi0   = s * sinv + half;             // yi(t) = yi0 + t*cos

    v8f c = {};  // 16x16 fp32 accumulator (8 VGPRs)

    // fp32 A(16x4) path: lane L holds M=L&15, K = 2*(L>>4)+{0,1}
    v2f bones; bones.x = 1.0f; bones.y = 1.0f;       // all-ones B => row-sum
    for (int t0 = 0; t0 < IMG_N; t0 += 4) {
        float ta = (float)(t0 + 2 * h) - half;
        float tb = ta + 1.0f;
        v2f aa;
        aa.x = sample_bilinear(simg, xi0 - ta * sinv, yi0 + ta * cosv);
        aa.y = sample_bilinear(simg, xi0 - tb * sinv, yi0 + tb * cosv);
        c = __builtin_amdgcn_wmma_f32_16x16x4_f32(
                /*neg_a=*/false, aa, /*neg_b=*/false, bones,
                /*c_mod=*/(short)0, c, /*reuse_a=*/false, /*reuse_b=*/false);
    }

    // D column n=0 lives in lanes 0 (M=0..7) and 16 (M=8..15), VGPR r = M row.
    if ((lane & 15) == 0) {
        const int mb = h * 8;
        const size_t ob = ((size_t)b * NANG + a) * (size_t)NPOSN + (size_t)(pbase + mb);
        #pragma unroll
        for (int r = 0; r < 8; ++r) out[ob + r] = c[r];
    }
}

extern "C" void kernel_launch(void* const* d_in, const int* in_sizes, int n_in,
                              void* d_out, int out_size, void* d_ws, size_t ws_size,
                              hipStream_t stream) {
    (void)in_sizes; (void)n_in; (void)out_size; (void)d_ws; (void)ws_size;
    const float* image     = (const float*)d_in[0];
    const float* thetas    = (const float*)d_in[1];
    const float* positions = (const float*)d_in[2];
    float*       out       = (float*)d_out;

    const dim3 grid(NBATCH * NANG * 2);  // 1440 blocks
    const dim3 block(256);               // 8 wave32s
    const size_t lds_bytes = (size_t)IMG_N * IMG_N * sizeof(float);  // 256 KB

    hipLaunchKernelGGL(radon_fwd_kernel, grid, block, lds_bytes, stream,
                       image, thetas, positions, out);
}